// CifLayer_88450556494218
// MI455X (gfx1250) — compile-verified
//
#include <hip/hip_runtime.h>
#include <hip/hip_bf16.h>

typedef __attribute__((ext_vector_type(2))) float v2f;
typedef __attribute__((ext_vector_type(4))) float v4f;
typedef __attribute__((ext_vector_type(8))) float v8f;

constexpr int B_ = 16;
constexpr int T_ = 4096;
constexpr int D_ = 512;
constexpr int U_ = 128;
constexpr int CH = 256;          // t-chunk for acoustic accumulation

// ---------------------------------------------------------------------------
// K0: zero acoustic output + per-batch alpha sums
// ---------------------------------------------------------------------------
__global__ void cif_zero_kernel(float* __restrict__ acoustic, float* __restrict__ sum_a) {
    size_t i = (size_t)blockIdx.x * blockDim.x + threadIdx.x;
    size_t n = (size_t)B_ * U_ * D_;
    if (i < n) acoustic[i] = 0.0f;
    if (i < (size_t)B_) sum_a[i] = 0.0f;
}

// ---------------------------------------------------------------------------
// K1: depthwise conv3 + residual + relu, then dot with lin_w via WMMA f32
//     one wave handles 16 timesteps (M dim); K-loop over D in steps of 4.
//     A[m,k] = relu(conv(x))[t0+m, d+k];  B[k,0] = lin_w[d+k], else 0.
//     Boundary (t=0 / t=T-1) handled branch-free: clamped row pointer +
//     0/1 lane mask multiplied into the halo vectors, so EXEC stays fully
//     converged through the whole K-loop (WMMA requires EXEC all-ones).
// ---------------------------------------------------------------------------
__global__ __launch_bounds__(256) void cif_alpha_wmma_kernel(
    const float* __restrict__ enc, const int* __restrict__ enc_len,
    const float* __restrict__ conv_w, const float* __restrict__ conv_b,
    const float* __restrict__ lin_w, const float* __restrict__ lin_b,
    float* __restrict__ a_raw, float* __restrict__ sum_a) {

    __shared__ float s_cw[D_ * 3];
    __shared__ float s_cb[D_];
    __shared__ float s_lw[D_];

    for (int i = threadIdx.x; i < D_ * 3; i += 256) s_cw[i] = conv_w[i];
    for (int i = threadIdx.x; i < D_; i += 256) { s_cb[i] = conv_b[i]; s_lw[i] = lin_w[i]; }
    __syncthreads();

    const int wave = threadIdx.x >> 5;
    const int lane = threadIdx.x & 31;
    const int gw   = blockIdx.x * 8 + wave;     // 0 .. B*T/16-1 (4096 waves)
    const int b    = gw >> 8;                   // T/16 = 256 tiles per batch
    const int t0   = (gw & 255) << 4;

    const int  r    = lane & 15;
    const int  t    = t0 + r;                   // row this lane feeds into A
    const bool hi   = (lane >= 16);             // holds K=2,3 instead of K=0,1
    const float* xrow = enc + ((size_t)b * T_ + t) * D_;

    // clamped halo rows + 0/1 masks (branch-free boundary handling)
    const float mMask = (t > 0)      ? 1.0f : 0.0f;
    const float pMask = (t < T_ - 1) ? 1.0f : 0.0f;
    const float* xmrow = xrow - ((t > 0)      ? D_ : 0);
    const float* xprow = xrow + ((t < T_ - 1) ? D_ : 0);

    v8f acc;
#pragma unroll
    for (int i = 0; i < 8; ++i) acc[i] = 0.0f;

#pragma unroll 2
    for (int d = 0; d < D_; d += 4) {
        v4f xc = *(const v4f*)(xrow  + d);
        v4f xm = *(const v4f*)(xmrow + d);
        v4f xp = *(const v4f*)(xprow + d);

        float y[4];
#pragma unroll
        for (int j = 0; j < 4; ++j) {
            const int dj = d + j;
            float v = fmaf(s_cw[dj * 3 + 0], xm[j] * mMask,
                      fmaf(s_cw[dj * 3 + 1], xc[j],
                      fmaf(s_cw[dj * 3 + 2], xp[j] * pMask, s_cb[dj])));
            v += xc[j];                          // residual
            y[j] = fmaxf(v, 0.0f);               // relu
        }

        // A fragment: lanes 0-15 -> K = d+0,d+1 ; lanes 16-31 -> K = d+2,d+3
        v2f A;
        A.x = hi ? y[2] : y[0];
        A.y = hi ? y[3] : y[1];

        // B fragment: only column N=0 carries lin_w
        v4f lw4 = *(const v4f*)(s_lw + d);
        v2f Bf;
        Bf.x = (lane == 0) ? lw4[0] : (lane == 16) ? lw4[2] : 0.0f;
        Bf.y = (lane == 0) ? lw4[1] : (lane == 16) ? lw4[3] : 0.0f;

        acc = __builtin_amdgcn_wmma_f32_16x16x4_f32(
            false, A, false, Bf, (short)0, acc, false, false);
    }

    // Column N=0 of D: lane 0 holds M=0..7, lane 16 holds M=8..15
    if (lane == 0 || lane == 16) {
        const int base_m = (lane == 0) ? 0 : 8;
        const int len = enc_len[b];
        const float lb = lin_b[0];
        float lsum = 0.0f;
#pragma unroll
        for (int i = 0; i < 8; ++i) {
            const int tt = t0 + base_m + i;
            float logit = acc[i] + lb;
            float a = 1.0f / (1.0f + expf(-logit));   // SMOOTH=1, NOISE=0 -> relu(sigmoid)=sigmoid
            a = (tt < len) ? a : 0.0f;
            a_raw[(size_t)b * T_ + tt] = a;
            lsum += a;
        }
        atomicAdd(&sum_a[b], lsum);
    }
}

// ---------------------------------------------------------------------------
// K2: per-batch parallel scan. integrate_post(t) = S_t - floor(S_{t-1});
//     fires when >= 1.  Produces fires, alphas, token_num_hat outputs and
//     per-t (primary weight, rem weight, bin|fired) for the acoustic pass.
// ---------------------------------------------------------------------------
__global__ __launch_bounds__(256) void cif_scan_kernel(
    const float* __restrict__ a_raw, const float* __restrict__ sum_a,
    const int* __restrict__ tgt_len,
    float* __restrict__ alphas_out, float* __restrict__ fires_out,
    float* __restrict__ tok_out,
    float* __restrict__ w_pri, float* __restrict__ w_rem,
    int* __restrict__ binf, int* __restrict__ f_total) {

    const int b = blockIdx.x;
    const int j = threadIdx.x;                 // 256 threads, 16 t's each
    __shared__ float part[256];

    const float sa    = sum_a[b];
    const float scale = (float)tgt_len[b] / sa;

    float loc[16];
    const size_t base = (size_t)b * T_ + (size_t)j * 16;
    float s = 0.0f;
#pragma unroll
    for (int k = 0; k < 16; ++k) { loc[k] = a_raw[base + k] * scale; s += loc[k]; }
    part[j] = s;
    __syncthreads();

    // inclusive Hillis-Steele scan over the 256 partial sums
    for (int off = 1; off < 256; off <<= 1) {
        float v = part[j];
        float add = (j >= off) ? part[j - off] : 0.0f;
        __syncthreads();
        part[j] = v + add;
        __syncthreads();
    }

    float S_prev = (j > 0) ? part[j - 1] : 0.0f;

#pragma unroll
    for (int k = 0; k < 16; ++k) {
        const float al   = loc[k];
        const float S    = S_prev + al;
        const float Fprv = floorf(S_prev);
        const float fire_v = S - Fprv;          // reference "fires" value
        const bool  fired  = (fire_v >= 1.0f);  // THRESHOLD = 1
        const int   cbin   = (int)Fprv;         // fires strictly before t
        const float cur = fired ? (1.0f - (S_prev - Fprv)) : al;
        const float rem = fired ? (al - cur) : 0.0f;

        alphas_out[base + k] = al;
        fires_out[base + k]  = fire_v;
        w_pri[base + k] = cur;
        w_rem[base + k] = rem;
        binf[base + k]  = cbin | (fired ? (1 << 30) : 0);
        S_prev = S;
    }

    if (j == 0)   tok_out[b] = sa;
    if (j == 255) f_total[b] = (int)floorf(S_prev);
}

// ---------------------------------------------------------------------------
// K3: acoustic accumulation. Block = (b, chunk of 256 t). Each thread owns
//     2 channels; register-accumulate the open bin, atomically flush on fire
//     and at chunk boundary (bins may straddle chunks).
// ---------------------------------------------------------------------------
__global__ __launch_bounds__(256) void cif_acoustic_kernel(
    const float* __restrict__ enc,
    const float* __restrict__ w_pri, const float* __restrict__ w_rem,
    const int* __restrict__ binf, const int* __restrict__ f_total,
    float* __restrict__ acoustic) {

    const int b     = blockIdx.x / (T_ / CH);
    const int chunk = blockIdx.x % (T_ / CH);
    const int t0    = chunk * CH;
    const int tid   = threadIdx.x;

    __shared__ float s_pri[CH];
    __shared__ float s_rem[CH];
    __shared__ int   s_bf[CH];

    const size_t sbase = (size_t)b * T_ + t0;
    s_pri[tid] = w_pri[sbase + tid];
    s_rem[tid] = w_rem[sbase + tid];
    s_bf[tid]  = binf[sbase + tid];
    __syncthreads();

    const int limit = min(U_, f_total[b]);      // only bins closed by a fire count
    const int d = tid * 2;
    const float* xb = enc + ((size_t)b * T_ + t0) * D_ + d;
    float* ab = acoustic + (size_t)b * U_ * D_ + d;

    float ax = 0.0f, ay = 0.0f;
    for (int k = 0; k < CH; ++k) {
        v2f x = *(const v2f*)(xb + (size_t)k * D_);
        const int bf    = s_bf[k];
        const int bin   = bf & 0x3FFFFFFF;
        const bool fired = (bf >> 30) != 0;
        const float pri = s_pri[k];
        ax = fmaf(pri, x.x, ax);
        ay = fmaf(pri, x.y, ay);
        if (fired) {
            if (bin < limit) {
                atomicAdd(&ab[(size_t)bin * D_ + 0], ax);
                atomicAdd(&ab[(size_t)bin * D_ + 1], ay);
            }
            const float rm = s_rem[k];
            ax = rm * x.x;
            ay = rm * x.y;
        }
    }
    // boundary flush: the still-open bin continues into the next chunk
    const int bfL = s_bf[CH - 1];
    const int endbin = (bfL & 0x3FFFFFFF) + ((bfL >> 30) != 0 ? 1 : 0);
    if (endbin < limit) {
        atomicAdd(&ab[(size_t)endbin * D_ + 0], ax);
        atomicAdd(&ab[(size_t)endbin * D_ + 1], ay);
    }
}

// ---------------------------------------------------------------------------
extern "C" void kernel_launch(void* const* d_in, const int* in_sizes, int n_in,
                              void* d_out, int out_size, void* d_ws, size_t ws_size,
                              hipStream_t stream) {
    const float* enc     = (const float*)d_in[0];   // (B,T,D)
    const int*   enc_len = (const int*)d_in[1];     // (B,)
    const int*   tgt_len = (const int*)d_in[2];     // (B,)
    const float* conv_w  = (const float*)d_in[3];   // (D,1,KW)
    const float* conv_b  = (const float*)d_in[4];   // (D,)
    const float* lin_w   = (const float*)d_in[5];   // (1,D)
    const float* lin_b   = (const float*)d_in[6];   // (1,)

    float* out      = (float*)d_out;
    float* acoustic = out;                                   // B*U*D
    float* fires    = acoustic + (size_t)B_ * U_ * D_;       // B*T
    float* tok      = fires + (size_t)B_ * T_;               // B
    float* alphas   = tok + B_;                              // B*T

    float* ws      = (float*)d_ws;
    float* a_raw   = ws;                                     // B*T
    float* w_pri   = a_raw + (size_t)B_ * T_;                // B*T
    float* w_rem   = w_pri + (size_t)B_ * T_;                // B*T
    int*   binf    = (int*)(w_rem + (size_t)B_ * T_);        // B*T
    float* sum_a   = (float*)(binf + (size_t)B_ * T_);       // B
    int*   f_total = (int*)(sum_a + B_);                     // B

    cif_zero_kernel<<<(B_ * U_ * D_ + 255) / 256, 256, 0, stream>>>(acoustic, sum_a);

    // B*T/16 waves, 8 waves per block
    cif_alpha_wmma_kernel<<<(B_ * T_ / 16) / 8, 256, 0, stream>>>(
        enc, enc_len, conv_w, conv_b, lin_w, lin_b, a_raw, sum_a);

    cif_scan_kernel<<<B_, 256, 0, stream>>>(
        a_raw, sum_a, tgt_len, alphas, fires, tok, w_pri, w_rem, binf, f_total);

    cif_acoustic_kernel<<<B_ * (T_ / CH), 256, 0, stream>>>(
        enc, w_pri, w_rem, binf, f_total, acoustic);
}